// PoissonNLMDenoiser_61933428417258
// MI455X (gfx1250) — compile-verified
//
#include <hip/hip_runtime.h>
#include <stdint.h>

// ============================================================================
// Poisson NLM denoiser for MI455X (gfx1250, wave32)
//   out = clip( (NLM(2*sqrt(max(x,0)+3/8)) / 2)^2 - 3/8, 0, 1 )
// 11x11 pixel-wise similarity window, reflect borders.
//
// Roofline: 100 MB traffic (~4.3us @ 23.3 TB/s) vs 1.52G v_exp_f32 + ~6G VALU
// -> transcendental/VALU bound. Inner loop folded to 4 VALU + 1 TRANS per
// window offset via exp2(K*s^2 + c1*s + c0). Interior tiles staged to LDS via
// the Tensor Data Mover (tensor_load_to_lds + s_wait_tensorcnt); border tiles
// use per-lane reflected loads.
// ============================================================================

#define IMG_DIM  512
#define N_PLANES 48              // 16 batch * 3 channels

#define HALO     5               // search radius (11x11)
#define TILE_W   32
#define TILE_H   64
#define ROWS_PT  8               // rows per thread: 256 thr = 32x8, 8 rows each
#define SM_W     (TILE_W + 2*HALO)   // 42
#define SM_H     (TILE_H + 2*HALO)   // 74
#define SM_N     (SM_W * SM_H)       // 3108 floats = 12432 B LDS
#define NTHREADS 256

typedef unsigned int u32x4_t __attribute__((ext_vector_type(4)));
typedef int          i32x4_t __attribute__((ext_vector_type(4)));
typedef int          i32x8_t __attribute__((ext_vector_type(8)));
typedef __attribute__((address_space(3))) float lds_f32_t;

#if defined(__HIP_DEVICE_COMPILE__) && __has_builtin(__builtin_amdgcn_tensor_load_to_lds)
#define HAVE_TDM 1
#else
#define HAVE_TDM 0
#endif

__device__ __forceinline__ int reflect_idx(int v) {
  v = (v < 0) ? -v : v;
  return (v > IMG_DIM - 1) ? (2 * (IMG_DIM - 1) - v) : v;
}

__global__ __launch_bounds__(NTHREADS)
void nlm_anscombe_kernel(const float* __restrict__ x, float* __restrict__ out) {
  __shared__ float sm[SM_H][SM_W];
  float* smf = &sm[0][0];

  const int tid    = threadIdx.x;
  const int tileX0 = blockIdx.x * TILE_W;
  const int tileY0 = blockIdx.y * TILE_H;
  const float* __restrict__ src = x   + (size_t)blockIdx.z * (IMG_DIM * IMG_DIM);
  float*       __restrict__ dst = out + (size_t)blockIdx.z * (IMG_DIM * IMG_DIM);

  const bool interior =
      (tileX0 >= HALO) && (tileX0 + TILE_W + HALO <= IMG_DIM) &&
      (tileY0 >= HALO) && (tileY0 + TILE_H + HALO <= IMG_DIM);

  // ---- Stage raw tile (+halo) into LDS ------------------------------------
  // Issued first so the TDM descriptor build + tensor_load_to_lds sit at the
  // top of the emitted code (DMA starts before anything else happens).
#if HAVE_TDM
  if (interior) {
    if (tid < 32) {   // wave 0 issues the TDM DMA (EXEC-independent, per-wave)
      const uint64_t gaddr = (uint64_t)(uintptr_t)
          (src + (size_t)(tileY0 - HALO) * IMG_DIM + (tileX0 - HALO));
      const uint32_t laddr = (uint32_t)(uintptr_t)(lds_f32_t*)smf;

      // D# group 0: count=1, lds_addr, global_addr[56:0], type=2
      u32x4_t g0;
      g0[0] = 1u;
      g0[1] = laddr;
      g0[2] = (uint32_t)gaddr;
      g0[3] = (uint32_t)(gaddr >> 32) | (2u << 30);

      // D# group 1: data_size=4B, tensor 42x74 (no OOB inside tile),
      // tile 42x74, row stride 512 elements
      i32x8_t g1;
      g1[0] = (int)(2u << 16);                 // data_size = 2 (4 bytes)
      g1[1] = (int)((uint32_t)SM_W << 16);     // tensor_dim0 low16 @ bits63:48
      g1[2] = (int)((uint32_t)SM_H << 16);     // tensor_dim1 low16 @ bits95:80
      g1[3] = (int)((uint32_t)SM_W << 16);     // tile_dim0 @ bits127:112
      g1[4] = (int)SM_H;                       // tile_dim1 (tile_dim2 = 0)
      g1[5] = IMG_DIM;                         // tensor_dim0_stride low32
      g1[6] = 0;
      g1[7] = 0;

      const i32x4_t gz = {0, 0, 0, 0};
#if __clang_major__ >= 23
      const i32x8_t gz8 = {0, 0, 0, 0, 0, 0, 0, 0};
      __builtin_amdgcn_tensor_load_to_lds(g0, g1, gz, gz, gz8, 0);
#else
      __builtin_amdgcn_tensor_load_to_lds(g0, g1, gz, gz, 0);
#endif
      __builtin_amdgcn_s_wait_tensorcnt(0);
    }
  } else
#endif
  {
    if (interior) {
      const float* tsrc = src + (size_t)(tileY0 - HALO) * IMG_DIM + (tileX0 - HALO);
      for (int idx = tid; idx < SM_N; idx += NTHREADS) {
        const int r = idx / SM_W;
        const int c = idx - r * SM_W;
        smf[idx] = tsrc[r * IMG_DIM + c];
      }
    } else {
      for (int idx = tid; idx < SM_N; idx += NTHREADS) {
        const int r  = idx / SM_W;
        const int c  = idx - r * SM_W;
        const int gy = reflect_idx(tileY0 - HALO + r);
        const int gx = reflect_idx(tileX0 - HALO + c);
        smf[idx] = src[gy * IMG_DIM + gx];
      }
    }
  }
  __syncthreads();

  // ---- In-place Anscombe transform: a = 2*sqrt(max(v,0) + 3/8) ------------
  for (int idx = tid; idx < SM_N; idx += NTHREADS) {
    const float v = fmaxf(smf[idx], 0.0f);
    smf[idx] = 2.0f * __builtin_amdgcn_sqrtf(v + 0.375f);
  }
  __syncthreads();

  // ---- NLM accumulation ----------------------------------------------------
  // exp(-(a-s)^2/D) = exp2(K*(a-s)^2), K = -log2(e)/D, D = 0.01*9 + 1e-10
  // expanded: exp2( s*(s*K + c1) + c0 ),  c1 = -2*K*a,  c0 = K*a^2
  const float K = -1.44269504088896340736f / (0.1f * 0.1f * 9.0f + 1e-10f);

  const int tx   = tid & (TILE_W - 1);   // 0..31 (lane within wave)
  const int ty   = tid >> 5;             // 0..7  (wave id)
  const int row0 = ty * ROWS_PT;         // first tile-local output row

  float c0[ROWS_PT], c1[ROWS_PT], acc[ROWS_PT], ws[ROWS_PT];
#pragma unroll
  for (int k = 0; k < ROWS_PT; ++k) {
    const float a = sm[row0 + k + HALO][tx + HALO];
    c1[k]  = -2.0f * K * a;
    c0[k]  = K * a * a;
    acc[k] = 0.0f;
    ws[k]  = 0.0f;
  }

#pragma unroll 1
  for (int dj = 0; dj <= 2 * HALO; ++dj) {
    // 18 LDS values feed 8 rows x 11 vertical offsets (register rolling)
    float col[ROWS_PT + 2 * HALO];
#pragma unroll
    for (int r = 0; r < ROWS_PT + 2 * HALO; ++r)
      col[r] = sm[row0 + r][tx + dj];

#pragma unroll
    for (int k = 0; k < ROWS_PT; ++k) {
#pragma unroll
      for (int di = 0; di <= 2 * HALO; ++di) {
        const float s   = col[k + di];
        const float t   = __builtin_fmaf(s, K, c1[k]);
        const float arg = __builtin_fmaf(s, t, c0[k]);
        const float w   = __builtin_amdgcn_exp2f(arg);   // v_exp_f32 (TRANS)
        acc[k] = __builtin_fmaf(w, s, acc[k]);
        ws[k] += w;
      }
    }
  }

  // ---- Inverse Anscombe + clamp + store ------------------------------------
#pragma unroll
  for (int k = 0; k < ROWS_PT; ++k) {
    const float den = acc[k] * __builtin_amdgcn_rcpf(ws[k] + 1e-10f);
    const float q   = 0.5f * den;
    float o = __builtin_fmaf(q, q, -0.375f);
    o = fminf(fmaxf(o, 0.0f), 1.0f);
    dst[(tileY0 + row0 + k) * IMG_DIM + (tileX0 + tx)] = o;
  }
}

extern "C" void kernel_launch(void* const* d_in, const int* in_sizes, int n_in,
                              void* d_out, int out_size, void* d_ws, size_t ws_size,
                              hipStream_t stream) {
  (void)in_sizes; (void)n_in; (void)d_ws; (void)ws_size; (void)out_size;
  const float* x = (const float*)d_in[0];
  float* out     = (float*)d_out;

  dim3 grid(IMG_DIM / TILE_W, IMG_DIM / TILE_H, N_PLANES);  // 16 x 8 x 48
  nlm_anscombe_kernel<<<grid, dim3(NTHREADS), 0, stream>>>(x, out);
}